// IMUHOIModel_43782896615958
// MI455X (gfx1250) — compile-verified
//
#include <hip/hip_runtime.h>
#include <hip/hip_bf16.h>
#include <stdint.h>

// Problem constants (match the JAX reference)
#define B_ 1024
#define T_ 2048
#define CONTACT_THRESHOLD 0.5f
#define EPS_ 1e-6f
#define DEFAULT_LEN 0.1f

#define TT 8              // timesteps per tile
#define NT (T_ / TT)      // 256 tiles
#define EPW 32            // batch elements per workgroup == one wave32

// Dense LDS tile row lengths (floats per batch element per tile)
#define PROW 16           // TT*2
#define HROW 48           // TT*6
#define RROW 72           // TT*9
#define OROW 24           // TT*3

typedef unsigned int       u32;
typedef unsigned long long u64;
typedef u32 u32x4 __attribute__((ext_vector_type(4)));
typedef u32 u32x8 __attribute__((ext_vector_type(8)));

// ---------- CDNA5 TDM / async ops + split wait counters ----------
__device__ __forceinline__ void tdm_load_2d(u32x4 g0, u32x8 g1) {
  // 2D tensor: D# groups 2/3 omitted (NULL) -> 2-group form
  asm volatile("tensor_load_to_lds %0, %1" :: "s"(g0), "s"(g1) : "memory");
}
__device__ __forceinline__ void async_st128(const float* g, unsigned ldsOff) {
  asm volatile("global_store_async_from_lds_b128 %0, %1, off"
               :: "v"(g), "v"(ldsOff) : "memory");
}
// TENSORcnt is in-order per wave: <=3 means "previous tile's 3 loads landed"
__device__ __forceinline__ void wait_tensor_le3() {
  asm volatile("s_wait_tensorcnt 0x3" ::: "memory");
}
__device__ __forceinline__ void wait_tensor0() {
  asm volatile("s_wait_tensorcnt 0x0" ::: "memory");
}
// ASYNCcnt tracks only our stores (in-order w.r.t. stores): <=6 frees the
// two-tiles-old output staging buffer while last tile's 6 stores stay in flight
__device__ __forceinline__ void wait_async_le6() {
  asm volatile("s_wait_asynccnt 0x6" ::: "memory");
}
__device__ __forceinline__ void wait_ds0() {
  asm volatile("s_wait_dscnt 0x0" ::: "memory");
}

__device__ __forceinline__ unsigned lds_off(const float* p) {
  // addrspace(3) -> generic keeps the wave-relative LDS byte offset in the low 32 bits
  return (unsigned)(size_t)p;
}

// ---------- Tensor DMA Descriptor (D#) builders ----------
// Group 1 (256b): data_size=4B, 2D tile {tile_dim0=rowElems, tile_dim1=EPW},
// tensor dims match the tile (no OOB clipping), row stride in elements.
__device__ __forceinline__ u32x8 make_g1(u32 rowElems, u64 strideElems) {
  u32x8 g;
  g[0] = (2u << 16);                          // wg_mask=0 | data_size=2 (4B) | no pad/iterate
  g[1] = (rowElems & 0xFFFFu) << 16;          // [47:32]=atomic_addr=0 | [63:48]=tensor_dim0.lo16
  g[2] = (rowElems >> 16) | ((u32)EPW << 16); // [79:64]=tensor_dim0.hi16 | [95:80]=tensor_dim1.lo16
  g[3] = (rowElems & 0xFFFFu) << 16;          // [111:96]=tensor_dim1.hi16=0 | [127:112]=tile_dim0
  g[4] = (u32)EPW;                            // [143:128]=tile_dim1=32 | [159:144]=tile_dim2=0
  g[5] = (u32)(strideElems & 0xFFFFFFFFull);  // tensor_dim0_stride.lo32
  g[6] = (u32)((strideElems >> 32) & 0xFFFFull); // stride.hi16 | tensor_dim1_stride.lo16=0
  g[7] = 0u;                                  // tensor_dim1_stride.hi32=0
  return g;
}
// Group 0 (128b): count=1 | lds_addr | global tile address | type=2 ("image")
__device__ __forceinline__ u32x4 make_g0(u32 ldsByte, u64 gaddr) {
  u32x4 g;
  g[0] = 1u;                                  // count=1, user descriptor, gather off
  g[1] = ldsByte;                             // lds_addr
  g[2] = (u32)gaddr;                          // global_addr[31:0]
  g[3] = ((u32)(gaddr >> 32) & 0x01FFFFFFu) | (2u << 30); // global_addr[56:32] | type=2
  return g;
}

__global__ void __launch_bounds__(EPW)
IMUHOIModel_fk_scan_kernel(const float* __restrict__ prob,
                           const float* __restrict__ hand,
                           const float* __restrict__ rot,
                           const float* __restrict__ tinit,
                           float* __restrict__ out)
{
  __shared__ __align__(16) float sProb[2][EPW * PROW];  //  4.0 KB
  __shared__ __align__(16) float sHand[2][EPW * HROW];  // 12.0 KB
  __shared__ __align__(16) float sRot [2][EPW * RROW];  // 18.0 KB
  __shared__ __align__(16) float sOut [2][EPW * OROW];  //  6.0 KB

  const int lane  = threadIdx.x;          // one batch element per lane
  const int bBase = blockIdx.x * EPW;
  const int b     = bBase + lane;

  // Tile-invariant D# group-1 descriptors (wave-uniform SGPR values)
  const u32x8 g1p = make_g1(PROW, (u64)T_ * 2ull);
  const u32x8 g1h = make_g1(HROW, (u64)T_ * 6ull);
  const u32x8 g1r = make_g1(RROW, (u64)T_ * 9ull);

  // LDS byte offsets of the staging buffers (loop-invariant)
  const u32 lp0 = lds_off(&sProb[0][0]), lp1 = lds_off(&sProb[1][0]);
  const u32 lh0 = lds_off(&sHand[0][0]), lh1 = lds_off(&sHand[1][0]);
  const u32 lr0 = lds_off(&sRot [0][0]), lr1 = lds_off(&sRot [1][0]);

  // Running global tile addresses (bytes); bump by constant strides per tile
  u64 gp = (u64)(size_t)prob + (u64)bBase * (T_ * 2ull * 4ull);
  u64 gh = (u64)(size_t)hand + (u64)bBase * (T_ * 6ull * 4ull);
  u64 gr = (u64)(size_t)rot  + (u64)bBase * (T_ * 9ull * 4ull);

  // --- scan state ---
  float px = tinit[b * 3 + 0];
  float py = tinit[b * 3 + 1];
  float pz = tinit[b * 3 + 2];
  int   cur  = -1;
  float dlx = 0.f, dly = 0.f, dlz = 0.f, dist = 0.f;
  bool  plc = false, prc = false;

  // prologue: TDM-prefetch tile 0 into buffer 0
  tdm_load_2d(make_g0(lp0, gp), g1p);
  tdm_load_2d(make_g0(lh0, gh), g1h);
  tdm_load_2d(make_g0(lr0, gr), g1r);
  gp += TT * 2 * 4; gh += TT * 6 * 4; gr += TT * 9 * 4;

  for (int tile = 0; tile < NT; ++tile) {
    const int buf = tile & 1;

    // issue tile k+1's DMA into the other buffer, then wait for tile k only
    if (tile + 1 < NT) {
      tdm_load_2d(make_g0(buf ? lp0 : lp1, gp), g1p);
      tdm_load_2d(make_g0(buf ? lh0 : lh1, gh), g1h);
      tdm_load_2d(make_g0(buf ? lr0 : lr1, gr), g1r);
      gp += TT * 2 * 4; gh += TT * 6 * 4; gr += TT * 9 * 4;
      wait_tensor_le3();          // tile k resident; k+1 still streaming
    } else {
      wait_tensor0();             // last tile: nothing else in flight
    }
    wait_async_le6();             // two-tiles-old stores retired -> sOut[buf] free

    const float* sP = &sProb[buf][lane * PROW];
    const float* sH = &sHand[buf][lane * HROW];
    const float* sR = &sRot [buf][lane * RROW];
    float*       sO = &sOut [buf][lane * OROW];

    #pragma unroll
    for (int k = 0; k < TT; ++k) {
      const int t = tile * TT + k;
      const float p0 = sP[k * 2 + 0];
      const float p1 = sP[k * 2 + 1];

      const bool lc = (t > 0) && (p0 > CONTACT_THRESHOLD);
      const bool rc = (t > 0) && (p1 > CONTACT_THRESHOLD);
      const bool ls = lc && !plc;
      const bool rs = rc && !prc;
      plc = lc; prc = rc;

      const int  nw    = ls ? 0 : (rs ? 1 : -1);       // left-start has priority
      const bool has_c = (cur == 0) ? lc : ((cur == 1) ? rc : false);
      const bool has_o = (cur == 0) ? rc : ((cur == 1) ? lc : false);
      const bool sw    = (nw == -1) && !has_c && has_o;            // handover
      const bool rel   = (nw == -1) && !sw && !has_c && (cur != -1);
      const bool regrab = (nw != -1) || sw;
      const int  curn  = (nw != -1) ? nw : (sw ? (1 - cur) : (rel ? -1 : cur));

      const float* Rm = sR + k * 9;

      if (regrab) {
        int g1 = 1 - cur; g1 = g1 < 0 ? 0 : (g1 > 1 ? 1 : g1);
        const int grab = (nw != -1) ? nw : g1;
        const float* hg = sH + k * 6 + grab * 3;
        const float vx = px - hg[0], vy = py - hg[1], vz = pz - hg[2];
        float d0 = sqrtf(vx * vx + vy * vy + vz * vz);
        const bool ok = d0 > EPS_;
        const float inv = ok ? (1.0f / d0) : 0.0f;
        const float ux = ok ? vx * inv : 0.0f;
        const float uy = ok ? vy * inv : 0.0f;
        const float uz = ok ? vz * inv : 1.0f;
        d0 = ok ? d0 : DEFAULT_LEN;
        // dir_local = R^T * unit
        dlx = Rm[0] * ux + Rm[3] * uy + Rm[6] * uz;
        dly = Rm[1] * ux + Rm[4] * uy + Rm[7] * uz;
        dlz = Rm[2] * ux + Rm[5] * uy + Rm[8] * uz;
        dist = d0;
      }
      cur = curn;
      if (cur != -1) {
        const float* hc = sH + k * 6 + cur * 3;
        // dir_world = R * dir_local
        const float wx = Rm[0] * dlx + Rm[1] * dly + Rm[2] * dlz;
        const float wy = Rm[3] * dlx + Rm[4] * dly + Rm[5] * dlz;
        const float wz = Rm[6] * dlx + Rm[7] * dly + Rm[8] * dlz;
        px = hc[0] + wx * dist;
        py = hc[1] + wy * dist;
        pz = hc[2] + wz * dist;
      }
      sO[k * 3 + 0] = px;
      sO[k * 3 + 1] = py;
      sO[k * 3 + 2] = pz;
    }

    // flush this tile's outputs: LDS -> global, coalesced async 16B chunks.
    // These overlap the next tile's compute; final S_ENDPGM wait-idles them.
    wait_ds0();  // make the ds_stores to sOut[buf] visible to the async engine
    #pragma unroll
    for (int it = 0; it < 6; ++it) {
      int i = it * 32 + lane;
      int e = i / 6, c = i % 6;
      float* g = out + ((size_t)(bBase + e) * T_ + tile * TT) * 3 + c * 4;
      async_st128(g, lds_off(&sOut[buf][0] + e * OROW + c * 4));
    }
  }
}

extern "C" void kernel_launch(void* const* d_in, const int* in_sizes, int n_in,
                              void* d_out, int out_size, void* d_ws, size_t ws_size,
                              hipStream_t stream) {
  (void)in_sizes; (void)n_in; (void)out_size; (void)d_ws; (void)ws_size;
  const float* prob  = (const float*)d_in[0];  // [B,T,2]
  const float* hand  = (const float*)d_in[1];  // [B,T,2,3]
  const float* rot   = (const float*)d_in[2];  // [B,T,3,3]
  const float* tinit = (const float*)d_in[3];  // [B,3]
  float* out = (float*)d_out;                  // [B,T,3]

  hipLaunchKernelGGL(IMUHOIModel_fk_scan_kernel,
                     dim3(B_ / EPW), dim3(EPW), 0, stream,
                     prob, hand, rot, tinit, out);
}